// LoopModel_54142357733839
// MI455X (gfx1250) — compile-verified
//
#include <hip/hip_runtime.h>
#include <hip/hip_bf16.h>

typedef float v2f __attribute__((ext_vector_type(2)));
typedef float v8f __attribute__((ext_vector_type(8)));

#define E_EDGES   512
#define C_CH      64
#define HW        784          // 28*28
#define CP        (C_CH * HW)  // 50176 floats per edge / corner
#define NCORN     256
#define NPAIRS    (2 * E_EDGES)

// ---------------------------------------------------------------------------
// Kernel 1: build corner->edge CSR (counts, offsets, edge list). 1 block.
// pairs: (2E, 2) int32 [corner, edge]
// ---------------------------------------------------------------------------
__global__ __launch_bounds__(256) void build_csr_kernel(
    const int* __restrict__ pairs, int* __restrict__ counts,
    int* __restrict__ offsets, int* __restrict__ list) {
  __shared__ int scnt[NCORN];
  __shared__ int soff[NCORN];
  const int c = threadIdx.x;  // one thread per corner
  int cnt = 0;
  for (int p = 0; p < NPAIRS; ++p)
    if (pairs[2 * p] == c) ++cnt;
  scnt[c] = cnt;
  counts[c] = cnt;
  __syncthreads();
  if (c == 0) {
    int acc = 0;
    for (int i = 0; i < NCORN; ++i) { soff[i] = acc; acc += scnt[i]; }
  }
  __syncthreads();
  offsets[c] = soff[c];
  int idx = soff[c];
  for (int p = 0; p < NPAIRS; ++p)
    if (pairs[2 * p] == c) list[idx++] = pairs[2 * p + 1];
}

// ---------------------------------------------------------------------------
// Kernel 2: corner_features[c, j] = (1/max(deg,1)) * sum_{e in list(c)} x[e, j]
// grid = (49, 256); 256 threads * float4 = 1024 floats/block; 49*1024 = CP
// ---------------------------------------------------------------------------
__global__ __launch_bounds__(256) void corner_feat_kernel(
    const float* __restrict__ x, const int* __restrict__ counts,
    const int* __restrict__ offsets, const int* __restrict__ list,
    float* __restrict__ cf) {
  const int c = blockIdx.y;
  const int j = (blockIdx.x * 256 + threadIdx.x) * 4;
  const int deg = counts[c];
  const int off = offsets[c];
  float4 acc = make_float4(0.f, 0.f, 0.f, 0.f);
  for (int i = 0; i < deg; ++i) {
    const int e = list[off + i];
    const float4 v = *(const float4*)&x[(size_t)e * CP + j];
    acc.x += v.x; acc.y += v.y; acc.z += v.z; acc.w += v.w;
  }
  const float inv = 1.0f / fmaxf((float)deg, 1.0f);
  acc.x *= inv; acc.y *= inv; acc.z *= inv; acc.w *= inv;
  *(float4*)&cf[(size_t)c * CP + j] = acc;
}

// ---------------------------------------------------------------------------
// Kernel 3: g[j] = max_e x[e, j].  grid = CP/256 = 196 blocks.
// ---------------------------------------------------------------------------
__global__ __launch_bounds__(256) void global_max_kernel(
    const float* __restrict__ x, float* __restrict__ g) {
  const int j = blockIdx.x * 256 + threadIdx.x;
  float m = -__builtin_inff();
  for (int e = 0; e < E_EDGES; ++e)
    m = fmaxf(m, x[(size_t)e * CP + j]);
  g[j] = m;
}

// ---------------------------------------------------------------------------
// Kernel 4: gout[o, p] = sum_c W_agg[o, 192 + c] * g[c, p]   (edge-invariant)
// grid = 196 blocks of 256; idx = o*HW + p
// ---------------------------------------------------------------------------
__global__ __launch_bounds__(256) void gout_kernel(
    const float* __restrict__ Wagg, const float* __restrict__ g,
    float* __restrict__ gout) {
  const int idx = blockIdx.x * 256 + threadIdx.x;  // < 64*784
  const int o = idx / HW;
  const int p = idx - o * HW;
  float acc = 0.f;
  for (int c = 0; c < C_CH; ++c)
    acc += Wagg[o * 256 + 192 + c] * g[c * HW + p];
  gout[idx] = acc;
}

// ---------------------------------------------------------------------------
// Kernel 5: main fused GEMM with V_WMMA_F32_16X16X4_F32.
// out[e,o,p] = relu( gout[o,p] + sum over 3 sources s (x, left, right):
//                    sum_c W_s[o,c] * src_s[e][c,p] )
// grid = (7 strip-chunks, 512 edges), block = 224 threads = 7 waves,
// each wave computes full M=64 x one 16-pixel strip (4 x v8f accumulators).
// W1..W3 staged in LDS (48 KB); A-fragments read as ds_load_b64.
// ---------------------------------------------------------------------------
__global__ __launch_bounds__(224) void edge_gemm_kernel(
    const float* __restrict__ x, const float* __restrict__ Wagg,
    const int* __restrict__ edge_corner, const float* __restrict__ cf,
    const float* __restrict__ gout, float* __restrict__ out) {
  __shared__ float Wlds[3 * 64 * 64];  // [src][o][c], 48 KB

  const int e = blockIdx.y;
  const int wave = threadIdx.x >> 5;            // 0..6
  const int strip = blockIdx.x * 7 + wave;      // 0..48
  const int lane = threadIdx.x & 31;
  const int half = lane >> 4;                   // 0: lanes 0-15, 1: lanes 16-31
  const int l15 = lane & 15;

  // Stage W blocks for sources 0..2 (c-offsets 0, 64, 128) into LDS.
  for (int i = threadIdx.x; i < 3 * 64 * 64; i += 224) {
    const int s = i >> 12;
    const int o = (i >> 6) & 63;
    const int c = i & 63;
    Wlds[i] = Wagg[o * 256 + s * 64 + c];
  }
  __syncthreads();

  const int p0 = strip * 16;
  const float* xe = x + (size_t)e * CP;
  const int ci0 = edge_corner[2 * e];
  const int ci1 = edge_corner[2 * e + 1];
  const float* Lp = cf + (size_t)ci0 * CP;
  const float* Rp = cf + (size_t)ci1 * CP;

  // C/D layout: VGPR r, lanes 0-15 -> M=r, lanes 16-31 -> M=r+8; N = lane&15.
  v8f acc[4];
#pragma unroll
  for (int t = 0; t < 4; ++t)
#pragma unroll
    for (int r = 0; r < 8; ++r)
      acc[t][r] = gout[(t * 16 + r + half * 8) * HW + p0 + l15];

  const float* srcs[3] = {xe, Lp, Rp};
#pragma unroll
  for (int s = 0; s < 3; ++s) {
    const float* __restrict__ src = srcs[s];
    const int wbase = s * 4096;
    for (int kk = 0; kk < 16; ++kk) {
      // K-step of 4: this lane-half covers K = 4*kk + 2*half + {0,1}
      const int c0 = kk * 4 + half * 2;
      // B-matrix 4x16 (K x N): v0 = B[K=c0][N=l15], v1 = B[K=c0+1][N=l15]
      v2f b;
      b.x = src[c0 * HW + p0 + l15];
      b.y = src[(c0 + 1) * HW + p0 + l15];
#pragma unroll
      for (int t = 0; t < 4; ++t) {
        // A-matrix 16x4 (M x K): lanes hold M=l15; v0=K[c0], v1=K[c0+1]
        const int abase = wbase + (t * 16 + l15) * 64 + c0;  // 8B aligned
        v2f a;
        a.x = Wlds[abase];
        a.y = Wlds[abase + 1];
        acc[t] = __builtin_amdgcn_wmma_f32_16x16x4_f32(
            false, a, false, b, (short)0, acc[t], false, false);
      }
    }
  }

  float* oute = out + (size_t)e * CP;
#pragma unroll
  for (int t = 0; t < 4; ++t)
#pragma unroll
    for (int r = 0; r < 8; ++r) {
      const float v = acc[t][r];
      oute[(t * 16 + r + half * 8) * HW + p0 + l15] = v > 0.f ? v : 0.f;
    }
}

// ---------------------------------------------------------------------------
// Host launcher
// inputs: 0:x (f32 E*C*H*W), 1:W_agg (f32 64x256), 2:corner_edge_pairs (i32 2E*2),
//         3:edge_corner (i32 E*2), 4:num_corners (scalar)
// ---------------------------------------------------------------------------
extern "C" void kernel_launch(void* const* d_in, const int* in_sizes, int n_in,
                              void* d_out, int out_size, void* d_ws, size_t ws_size,
                              hipStream_t stream) {
  const float* x     = (const float*)d_in[0];
  const float* Wagg  = (const float*)d_in[1];
  const int*   pairs = (const int*)d_in[2];
  const int*   ecorn = (const int*)d_in[3];
  float* out = (float*)d_out;

  // workspace layout (floats)
  float* cf    = (float*)d_ws;                 // 256 * 50176
  float* g     = cf + (size_t)NCORN * CP;      // 50176
  float* gout  = g + CP;                       // 50176
  int* counts  = (int*)(gout + CP);            // 256
  int* offsets = counts + NCORN;               // 256
  int* list    = offsets + NCORN;              // 1024

  build_csr_kernel<<<1, 256, 0, stream>>>(pairs, counts, offsets, list);
  corner_feat_kernel<<<dim3(49, NCORN), 256, 0, stream>>>(x, counts, offsets, list, cf);
  global_max_kernel<<<CP / 256, 256, 0, stream>>>(x, g);
  gout_kernel<<<CP / 256, 256, 0, stream>>>(Wagg, g, gout);
  edge_gemm_kernel<<<dim3(7, E_EDGES), 224, 0, stream>>>(x, Wagg, ecorn, cf, gout, out);
}